// FlexState_9191230013801
// MI455X (gfx1250) — compile-verified
//
#include <hip/hip_runtime.h>
#include <math.h>

typedef __attribute__((ext_vector_type(2))) float v2f;
typedef __attribute__((ext_vector_type(8))) float v8f;

#define BATCH   16
#define SEQ     1024
#define CDIM    256
#define NHEADS  4
#define HDIM    64
#define LKV     1024
#define NLAYERS 4

static __device__ __forceinline__ v8f wmma_f32(v2f a, v2f b, v8f c) {
  // D = A(16x4 f32) * B(4x16 f32) + C(16x16 f32)
  return __builtin_amdgcn_wmma_f32_16x16x4_f32(false, a, false, b, (short)0, c,
                                               false, false);
}

// ---------------------------------------------------------------------------
// Projection: y[b][o][s] = sum_c w[o][c] * x[b][c][s]
// Workgroup: 8 waves -> 128(o) x 32(s) tile, K staged in LDS chunks of 64.
// ---------------------------------------------------------------------------
__global__ __launch_bounds__(256) void fs_proj_kernel(
    const float* __restrict__ x, const float* __restrict__ w,
    float* __restrict__ y) {
  __shared__ float w_s[128 * 65];  // [o 128][c 64], stride 65
  __shared__ float x_s[64 * 33];   // [c 64][s 32],  stride 33

  const int b    = blockIdx.x;
  const int sblk = blockIdx.y;   // 32 blocks of 32 columns
  const int oblk = blockIdx.z;   // 2 blocks of 128 rows
  const int tid  = threadIdx.x;
  const int lane = tid & 31;
  const int wv   = tid >> 5;
  const int n    = lane & 15;
  const int hh   = lane >> 4;

  const int obase = oblk * 128;
  const int sbase = sblk * 32;
  const float* xb = x + (size_t)b * CDIM * SEQ;

  v8f acc0, acc1;
  for (int j = 0; j < 8; ++j) { acc0[j] = 0.f; acc1[j] = 0.f; }

  for (int c0 = 0; c0 < CDIM; c0 += 64) {
    __syncthreads();
    for (int idx = tid; idx < 128 * 64; idx += 256) {
      int r = idx >> 6, cc = idx & 63;
      w_s[r * 65 + cc] = w[(size_t)(obase + r) * CDIM + c0 + cc];
    }
    for (int idx = tid; idx < 64 * 32; idx += 256) {
      int r = idx >> 5, cc = idx & 31;
      x_s[r * 33 + cc] = xb[(size_t)(c0 + r) * SEQ + sbase + cc];
    }
    __syncthreads();

#pragma unroll 4
    for (int k = 0; k < 64; k += 4) {
      const int ka = k + 2 * hh;
      v2f a;
      a.x = w_s[(wv * 16 + n) * 65 + ka];
      a.y = w_s[(wv * 16 + n) * 65 + ka + 1];
      v2f b0;
      b0.x = x_s[ka * 33 + n];
      b0.y = x_s[(ka + 1) * 33 + n];
      v2f b1;
      b1.x = x_s[ka * 33 + 16 + n];
      b1.y = x_s[(ka + 1) * 33 + 16 + n];
      acc0 = wmma_f32(a, b0, acc0);
      acc1 = wmma_f32(a, b1, acc1);
    }
  }

  float* yb = y + (size_t)b * CDIM * SEQ;
#pragma unroll
  for (int j = 0; j < 8; ++j) {
    const int row = obase + wv * 16 + j + 8 * hh;
    yb[(size_t)row * SEQ + sbase + n]      = acc0[j];
    yb[(size_t)row * SEQ + sbase + 16 + n] = acc1[j];
  }
}

// ---------------------------------------------------------------------------
// Attention: per (b, h, s-block of 128). Flash-attention with online softmax
// in the WMMA C-layout. kv staged in LDS in 128-row chunks (padded stride 65).
// ---------------------------------------------------------------------------
__global__ __launch_bounds__(256) void fs_attn_kernel(
    const float* __restrict__ y, const float* __restrict__ kvl,
    float* __restrict__ dst) {
  __shared__ float kvs[128 * 65];  // [l 128][d 64], stride 65
  __shared__ float ps[8 * 256];    // per-wave 16x16 P transpose scratch

  const int b    = blockIdx.x;
  const int h    = blockIdx.y;
  const int sblk = blockIdx.z;
  const int tid  = threadIdx.x;
  const int lane = tid & 31;
  const int wv   = tid >> 5;
  const int n    = lane & 15;
  const int hh   = lane >> 4;
  const int s0   = sblk * 128 + wv * 16;

  const float* yb  = y + (size_t)b * CDIM * SEQ + (size_t)h * HDIM * SEQ;
  const float* kvh = kvl + (size_t)h * LKV * HDIM;

  // Preload Q tile (16 s x 64 d) into registers, pre-scaled by 1/sqrt(64).
  v2f qreg[16];
#pragma unroll
  for (int kc = 0; kc < 16; ++kc) {
    const int dd = kc * 4 + 2 * hh;
    v2f q;
    q.x = yb[(size_t)dd * SEQ + s0 + n] * 0.125f;
    q.y = yb[(size_t)(dd + 1) * SEQ + s0 + n] * 0.125f;
    qreg[kc] = q;
  }

  v8f oacc[4];
  float mrow[8], rsum[8];
#pragma unroll
  for (int t = 0; t < 4; ++t)
    for (int j = 0; j < 8; ++j) oacc[t][j] = 0.f;
#pragma unroll
  for (int j = 0; j < 8; ++j) { mrow[j] = -INFINITY; rsum[j] = 0.f; }

  float* pw = ps + wv * 256;

  for (int lc = 0; lc < 8; ++lc) {
    __syncthreads();
    const float* kvc = kvh + (size_t)lc * 128 * HDIM;
    for (int idx = tid; idx < 128 * 64; idx += 256) {
      int r = idx >> 6, cc = idx & 63;
      kvs[r * 65 + cc] = kvc[r * 64 + cc];
    }
    if (lc + 1 < 8)
      __builtin_prefetch(kvh + (size_t)(lc + 1) * 128 * HDIM + tid * 32, 0, 1);
    __syncthreads();

    for (int lt = 0; lt < 8; ++lt) {
      const int ll = lt * 16;
      // ---- scores tile: S = Q (16x64) * K^T (64x16) ----
      v8f sc;
      for (int j = 0; j < 8; ++j) sc[j] = 0.f;
#pragma unroll
      for (int kc = 0; kc < 16; ++kc) {
        const int dd = kc * 4 + 2 * hh;
        v2f bb;
        bb.x = kvs[(ll + n) * 65 + dd];
        bb.y = kvs[(ll + n) * 65 + dd + 1];
        sc = wmma_f32(qreg[kc], bb, sc);
      }
      // ---- online softmax (elementwise in C layout) ----
      float pj[8];
#pragma unroll
      for (int j = 0; j < 8; ++j) {
        float v = sc[j];
        v = fmaxf(v, __shfl_xor(v, 1, 16));
        v = fmaxf(v, __shfl_xor(v, 2, 16));
        v = fmaxf(v, __shfl_xor(v, 4, 16));
        v = fmaxf(v, __shfl_xor(v, 8, 16));
        const float mn    = fmaxf(mrow[j], v);
        const float alpha = __expf(mrow[j] - mn);
        mrow[j] = mn;
        float p = __expf(sc[j] - mn);
        float r = p;
        r += __shfl_xor(r, 1, 16);
        r += __shfl_xor(r, 2, 16);
        r += __shfl_xor(r, 4, 16);
        r += __shfl_xor(r, 8, 16);
        rsum[j] = rsum[j] * alpha + r;
#pragma unroll
        for (int t = 0; t < 4; ++t) oacc[t][j] *= alpha;
        pj[j] = p;
      }
      // ---- transpose P (C layout -> A layout) through per-wave LDS ----
#pragma unroll
      for (int j = 0; j < 8; ++j) pw[(j + 8 * hh) * 16 + n] = pj[j];
      asm volatile("s_wait_dscnt 0x0" ::: "memory");
      // ---- O += P (16x16) * V (16x64) ----
#pragma unroll
      for (int kc2 = 0; kc2 < 4; ++kc2) {
        const int la = kc2 * 4 + 2 * hh;
        v2f pa;
        pa.x = pw[n * 16 + la];
        pa.y = pw[n * 16 + la + 1];
#pragma unroll
        for (int dt = 0; dt < 4; ++dt) {
          v2f vb;
          vb.x = kvs[(ll + la) * 65 + dt * 16 + n];
          vb.y = kvs[(ll + la + 1) * 65 + dt * 16 + n];
          oacc[dt] = wmma_f32(pa, vb, oacc[dt]);
        }
      }
    }
  }

  // ---- normalize and store back in (b, c, s) layout ----
  float* db = dst + (size_t)b * CDIM * SEQ;
#pragma unroll
  for (int dt = 0; dt < 4; ++dt) {
#pragma unroll
    for (int j = 0; j < 8; ++j) {
      const int cidx = h * HDIM + dt * 16 + n;
      db[(size_t)cidx * SEQ + s0 + j + 8 * hh] = oacc[dt][j] / rsum[j];
    }
  }
}

// ---------------------------------------------------------------------------
extern "C" void kernel_launch(void* const* d_in, const int* in_sizes, int n_in,
                              void* d_out, int out_size, void* d_ws,
                              size_t ws_size, hipStream_t stream) {
  (void)in_sizes; (void)n_in; (void)out_size; (void)ws_size;
  const float* x_in = (const float*)d_in[0];
  // d_in[1] = length: only its shape (1024) matters -> L == LKV, unused here.
  const float* w  = (const float*)d_in[2];
  const float* kv = (const float*)d_in[3];
  float* out = (float*)d_out;
  float* y   = (float*)d_ws;  // 16 MB projection buffer

  const dim3 proj_grid(BATCH, SEQ / 32, CDIM / 128);
  const dim3 attn_grid(BATCH, NHEADS, SEQ / 128);

  for (int i = 0; i < NLAYERS; ++i) {
    const float* src = (i == 0) ? x_in : out;  // d_out doubles as ping buffer
    fs_proj_kernel<<<proj_grid, 256, 0, stream>>>(
        src, w + (size_t)i * CDIM * CDIM, y);
    fs_attn_kernel<<<attn_grid, 256, 0, stream>>>(
        y, kv + (size_t)i * NHEADS * LKV * HDIM, out);
  }
}